// SimpleGAT_26027501814350
// MI455X (gfx1250) — compile-verified
//
#include <hip/hip_runtime.h>

// Problem constants (match reference)
#define NN 20000
#define EE 320000
#define ET (EE + NN)   // edges incl. self loops = 340000
#define GG 64
#define HH 8
#define NEG_SLOPE 0.2f
#define GN_EPS 1e-5f

typedef __attribute__((ext_vector_type(2))) float v2f;
typedef __attribute__((ext_vector_type(8))) float v8f;

static inline int cdiv(long a, long b) { return (int)((a + b - 1) / b); }

// ---------- helpers ----------

// Monotone order-preserving float->uint encoding for segment-max via atomicMax(u32)
__device__ __forceinline__ unsigned fenc(float f) {
  unsigned u = __float_as_uint(f);
  return (u & 0x80000000u) ? ~u : (u | 0x80000000u);
}
__device__ __forceinline__ float fdec(unsigned u) {
  unsigned v = (u & 0x80000000u) ? (u & 0x7fffffffu) : ~u;
  return __uint_as_float(v);
}

__device__ __forceinline__ void edge_sd(const int* __restrict__ ei, int e, int& s, int& d) {
  if (e < EE) { s = ei[e]; d = ei[EE + e]; }
  else { s = e - EE; d = s; }   // appended self loops
}

// ---------- fills / inits ----------

__global__ void fill_f32(float* p, float v, int n) {
  int i = blockIdx.x * blockDim.x + threadIdx.x;
  if (i < n) p[i] = v;
}
__global__ void fill_u32(unsigned* p, unsigned v, int n) {
  int i = blockIdx.x * blockDim.x + threadIdx.x;
  if (i < n) p[i] = v;
}
__global__ void init_rows_bias(float* __restrict__ out, const float* __restrict__ bias,
                               int F, int n) {
  int i = blockIdx.x * blockDim.x + threadIdx.x;
  if (i < n) out[i] = bias[i % F];
}
__global__ void count_nodes(const int* __restrict__ batch, float* __restrict__ cnt) {
  int i = blockIdx.x * blockDim.x + threadIdx.x;
  if (i < NN) atomicAdd(&cnt[batch[i]], 1.0f);
}

// ---------- WMMA GEMM: C[NN x F] = X[NN x K] @ W[K x F] + bias ----------
// One wave per 16x16 tile; V_WMMA_F32_16X16X4_F32 (fp32 in/out, K step 4).
// A 16x4 layout (ISA 7.12.2): lanes 0-15 hold K={0,1}, lanes 16-31 K={2,3},
// striped over 2 VGPRs; B mirrored with lane = column; C/D 8 VGPRs,
// row = r + 8*laneHalf, col = lane&15.
template<int K, int F>
__global__ void gemm_bias_wmma(const float* __restrict__ X, const float* __restrict__ W,
                               const float* __restrict__ B, float* __restrict__ C) {
  const int tilesN = F / 16;
  const int tiles = (NN / 16) * tilesN;
  int wid  = threadIdx.x >> 5;
  int tile = blockIdx.x * 8 + wid;        // wave-uniform guard -> EXEC all 1s for WMMA
  if (tile >= tiles) return;
  int tm = tile / tilesN, tn = tile % tilesN;
  int row0 = tm * 16, col0 = tn * 16;
  int lane = threadIdx.x & 31;
  int half = lane >> 4, l16 = lane & 15;
  v8f acc = {};
  #pragma unroll
  for (int k = 0; k < K; k += 4) {
    int ka = k + 2 * half;
    float ax = (ka     < K) ? X[(row0 + l16) * K + ka]     : 0.f;
    float ay = (ka + 1 < K) ? X[(row0 + l16) * K + ka + 1] : 0.f;
    float bx = (ka     < K) ? W[ka * F + col0 + l16]       : 0.f;
    float by = (ka + 1 < K) ? W[(ka + 1) * F + col0 + l16] : 0.f;
    v2f av = {ax, ay};
    v2f bv = {bx, by};
    acc = __builtin_amdgcn_wmma_f32_16x16x4_f32(false, av, false, bv,
                                                (short)0, acc, false, false);
  }
  float bias = B[col0 + l16];
  #pragma unroll
  for (int r = 0; r < 8; ++r)
    C[(row0 + r + 8 * half) * F + col0 + l16] = acc[r] + bias;
}

// ---------- GATv2 edge passes ----------

// Pass A: per-edge attention logits a[e,h] = sum_f leaky(xl[s,f]+xr[d,f]) * att_flat[f]
// (att flat [H*C] == feature index). Wave per edge; segment max via encoded atomicMax.
template<int F, int C>
__global__ void attn_logits(const float* __restrict__ xl, const float* __restrict__ xr,
                            const float* __restrict__ att, const int* __restrict__ ei,
                            float* __restrict__ abuf, unsigned* __restrict__ amax) {
  int wid  = threadIdx.x >> 5;
  int lane = threadIdx.x & 31;
  int e = blockIdx.x * 8 + wid;
  if (e >= ET) return;
  int s, d; edge_sd(ei, e, s, d);
  float part[HH];
  #pragma unroll
  for (int h = 0; h < HH; ++h) part[h] = 0.f;
  #pragma unroll
  for (int i = 0; i < F / 32; ++i) {
    int f = lane + 32 * i;
    float v = xl[s * F + f] + xr[d * F + f];
    v = (v > 0.f) ? v : NEG_SLOPE * v;
    float contrib = v * att[f];
    int h = f / C;
    #pragma unroll
    for (int hh = 0; hh < HH; ++hh) part[hh] += (hh == h) ? contrib : 0.f;
  }
  #pragma unroll
  for (int h = 0; h < HH; ++h) {
    float r = part[h];
    #pragma unroll
    for (int off = 16; off > 0; off >>= 1) r += __shfl_xor(r, off, 32);
    if (lane == 0) {
      abuf[e * HH + h] = r;
      atomicMax(&amax[d * HH + h], fenc(r));
    }
  }
}

// Pass B: exp(a - amax[dst]) and segment-sum
__global__ void attn_exp(float* __restrict__ abuf, const unsigned* __restrict__ amax,
                         float* __restrict__ asum, const int* __restrict__ ei) {
  int idx = blockIdx.x * blockDim.x + threadIdx.x;   // over ET*HH
  if (idx >= ET * HH) return;
  int e = idx >> 3, h = idx & 7;
  int s, d; edge_sd(ei, e, s, d);
  float m  = fdec(amax[d * HH + h]);
  float ae = __expf(abuf[idx] - m);
  abuf[idx] = ae;
  atomicAdd(&asum[d * HH + h], ae);
}

// Pass C: out[dst] += (a/asum[dst]) * xl[src]   (wave per edge, coalesced over F)
template<int F, int C>
__global__ void attn_aggregate(const float* __restrict__ xl, const float* __restrict__ abuf,
                               const float* __restrict__ asum, const int* __restrict__ ei,
                               float* __restrict__ out) {
  int wid  = threadIdx.x >> 5;
  int lane = threadIdx.x & 31;
  int e = blockIdx.x * 8 + wid;
  if (e >= ET) return;
  int s, d; edge_sd(ei, e, s, d);
  #pragma unroll
  for (int i = 0; i < F / 32; ++i) {
    int f = lane + 32 * i;
    int h = f / C;
    float w = abuf[e * HH + h] / asum[d * HH + h];
    atomicAdd(&out[d * F + f], w * xl[s * F + f]);
  }
}

// Layer-3 concat=False: mean over heads (+bias)
__global__ void head_mean(const float* __restrict__ agg64, const float* __restrict__ bias3,
                          float* __restrict__ h3) {
  int idx = blockIdx.x * blockDim.x + threadIdx.x;   // over NN*8
  if (idx >= NN * 8) return;
  int n = idx >> 3, c = idx & 7;
  float s = 0.f;
  #pragma unroll
  for (int h = 0; h < HH; ++h) s += agg64[n * 64 + h * 8 + c];
  h3[idx] = s * 0.125f + bias3[c];
}

// ---------- GraphNorm (3 passes) ----------

__global__ void gn_sum(const float* __restrict__ x, const int* __restrict__ batch,
                       float* __restrict__ gsum, int F) {
  int idx = blockIdx.x * blockDim.x + threadIdx.x;
  if (idx >= NN * F) return;
  int n = idx / F, f = idx % F;
  atomicAdd(&gsum[batch[n] * F + f], x[idx]);
}
__global__ void gn_center(const float* __restrict__ x, const int* __restrict__ batch,
                          const float* __restrict__ gsum, const float* __restrict__ cnt,
                          const float* __restrict__ ms, float* __restrict__ t,
                          float* __restrict__ gvar, int F) {
  int idx = blockIdx.x * blockDim.x + threadIdx.x;
  if (idx >= NN * F) return;
  int n = idx / F, f = idx % F;
  int g = batch[n];
  float c = fmaxf(cnt[g], 1.0f);
  float v = x[idx] - (gsum[g * F + f] / c) * ms[f];
  t[idx] = v;
  atomicAdd(&gvar[g * F + f], v * v);
}
__global__ void gn_final(const float* __restrict__ t, const int* __restrict__ batch,
                         const float* __restrict__ gvar, const float* __restrict__ cnt,
                         const float* __restrict__ w, const float* __restrict__ b,
                         float* __restrict__ out, int F) {
  int idx = blockIdx.x * blockDim.x + threadIdx.x;
  if (idx >= NN * F) return;
  int n = idx / F, f = idx % F;
  int g = batch[n];
  float c = fmaxf(cnt[g], 1.0f);
  float y = w[f] * t[idx] * rsqrtf(gvar[g * F + f] / c + GN_EPS) + b[f];
  out[idx] = fmaxf(y, 0.f);   // ReLU fused
}

// ---------- pooling + head ----------

__global__ void pool_sum(const float* __restrict__ h, const int* __restrict__ batch,
                         float* __restrict__ feat) {
  int idx = blockIdx.x * blockDim.x + threadIdx.x;   // over NN*8
  if (idx >= NN * 8) return;
  int n = idx >> 3, c = idx & 7;
  atomicAdd(&feat[batch[n] * 8 + c], h[idx]);
}
__global__ void pool_final(const float* __restrict__ featsum, const float* __restrict__ cnt,
                           const float* __restrict__ lin_w, const float* __restrict__ lin_b,
                           float* __restrict__ d_out) {
  int g = threadIdx.x;   // one block of GG threads
  if (g >= GG) return;
  float c = fmaxf(cnt[g], 1.0f);
  float feats[8];
  #pragma unroll
  for (int i = 0; i < 8; ++i) {
    feats[i] = featsum[g * 8 + i] / c;
    d_out[GG * 4 + g * 8 + i] = feats[i];          // features output
  }
  #pragma unroll
  for (int j = 0; j < 4; ++j) {
    float acc = lin_b[j];
    #pragma unroll
    for (int i = 0; i < 8; ++i) acc += feats[i] * lin_w[i * 4 + j];
    d_out[g * 4 + j] = acc;                        // logits output
  }
}

// ---------- driver ----------

extern "C" void kernel_launch(void* const* d_in, const int* in_sizes, int n_in,
                              void* d_out, int out_size, void* d_ws, size_t ws_size,
                              hipStream_t stream) {
  const float* x     = (const float*)d_in[0];
  const int*   ei    = (const int*)d_in[1];
  const int*   batch = (const int*)d_in[2];
  const float* W1l = (const float*)d_in[3];  const float* b1l = (const float*)d_in[4];
  const float* W1r = (const float*)d_in[5];  const float* b1r = (const float*)d_in[6];
  const float* att1 = (const float*)d_in[7]; const float* bias1 = (const float*)d_in[8];
  const float* gn1w = (const float*)d_in[9]; const float* gn1b = (const float*)d_in[10];
  const float* gn1ms = (const float*)d_in[11];
  const float* W2l = (const float*)d_in[12]; const float* b2l = (const float*)d_in[13];
  const float* W2r = (const float*)d_in[14]; const float* b2r = (const float*)d_in[15];
  const float* att2 = (const float*)d_in[16]; const float* bias2 = (const float*)d_in[17];
  const float* gn2w = (const float*)d_in[18]; const float* gn2b = (const float*)d_in[19];
  const float* gn2ms = (const float*)d_in[20];
  const float* W3l = (const float*)d_in[21]; const float* b3l = (const float*)d_in[22];
  const float* W3r = (const float*)d_in[23]; const float* b3r = (const float*)d_in[24];
  const float* att3 = (const float*)d_in[25]; const float* bias3 = (const float*)d_in[26];
  const float* gn3w = (const float*)d_in[27]; const float* gn3b = (const float*)d_in[28];
  const float* gn3ms = (const float*)d_in[29];
  const float* lin_w = (const float*)d_in[30]; const float* lin_b = (const float*)d_in[31];
  float* out = (float*)d_out;

  // workspace layout (floats); TMP aliases XR (XR dead after attn_logits)
  float* ws = (float*)d_ws;
  float*    XL   = ws + 0;                  // NN*256
  float*    XR   = ws + 5120000;            // NN*256 (also TMP for graphnorm)
  float*    TMP  = XR;
  float*    AGG  = ws + 10240000;           // NN*256
  float*    HIO  = ws + 15360000;           // NN*256 layer output / next input
  float*    ABUF = ws + 20480000;           // ET*8
  unsigned* AMAX = (unsigned*)(ws + 23200000); // NN*8
  float*    ASUM = ws + 23360000;           // NN*8
  float*    H3   = ws + 23520000;           // NN*8
  float*    H3GN = ws + 23680000;           // NN*8
  float*    GSUM = ws + 23840000;           // GG*256
  float*    GVAR = ws + 23856384;           // GG*256
  float*    CNT  = ws + 23872768;           // GG
  float*    FEAT = ws + 23872832;           // GG*8

  const int T = 256;
  const int eb  = cdiv(ET, 8);              // wave-per-edge blocks
  const int ehb = cdiv((long)ET * HH, T);

  // nodes-per-graph counts (shared by all GraphNorms + pooling)
  fill_f32<<<1, GG, 0, stream>>>(CNT, 0.f, GG);
  count_nodes<<<cdiv(NN, T), T, 0, stream>>>(batch, CNT);

  // ===== Layer 1: 19 -> 8x32 concat (F=256) =====
  {
    const int F = 256, C = 32;
    int gb = cdiv((NN / 16) * (F / 16), 8);
    gemm_bias_wmma<19, 256><<<gb, T, 0, stream>>>(x, W1l, b1l, XL);
    gemm_bias_wmma<19, 256><<<gb, T, 0, stream>>>(x, W1r, b1r, XR);
    fill_u32<<<cdiv(NN * 8, T), T, 0, stream>>>(AMAX, 0u, NN * 8);
    fill_f32<<<cdiv(NN * 8, T), T, 0, stream>>>(ASUM, 0.f, NN * 8);
    attn_logits<256, 32><<<eb, T, 0, stream>>>(XL, XR, att1, ei, ABUF, AMAX);
    attn_exp<<<ehb, T, 0, stream>>>(ABUF, AMAX, ASUM, ei);
    init_rows_bias<<<cdiv(NN * F, T), T, 0, stream>>>(AGG, bias1, F, NN * F);
    attn_aggregate<256, 32><<<eb, T, 0, stream>>>(XL, ABUF, ASUM, ei, AGG);
    fill_f32<<<cdiv(GG * F, T), T, 0, stream>>>(GSUM, 0.f, GG * F);
    fill_f32<<<cdiv(GG * F, T), T, 0, stream>>>(GVAR, 0.f, GG * F);
    gn_sum<<<cdiv(NN * F, T), T, 0, stream>>>(AGG, batch, GSUM, F);
    gn_center<<<cdiv(NN * F, T), T, 0, stream>>>(AGG, batch, GSUM, CNT, gn1ms, TMP, GVAR, F);
    gn_final<<<cdiv(NN * F, T), T, 0, stream>>>(TMP, batch, GVAR, CNT, gn1w, gn1b, HIO, F);
  }

  // ===== Layer 2: 256 -> 8x16 concat (F=128) =====
  {
    const int F = 128, C = 16;
    int gb = cdiv((NN / 16) * (F / 16), 8);
    gemm_bias_wmma<256, 128><<<gb, T, 0, stream>>>(HIO, W2l, b2l, XL);
    gemm_bias_wmma<256, 128><<<gb, T, 0, stream>>>(HIO, W2r, b2r, XR);
    fill_u32<<<cdiv(NN * 8, T), T, 0, stream>>>(AMAX, 0u, NN * 8);
    fill_f32<<<cdiv(NN * 8, T), T, 0, stream>>>(ASUM, 0.f, NN * 8);
    attn_logits<128, 16><<<eb, T, 0, stream>>>(XL, XR, att2, ei, ABUF, AMAX);
    attn_exp<<<ehb, T, 0, stream>>>(ABUF, AMAX, ASUM, ei);
    init_rows_bias<<<cdiv(NN * F, T), T, 0, stream>>>(AGG, bias2, F, NN * F);
    attn_aggregate<128, 16><<<eb, T, 0, stream>>>(XL, ABUF, ASUM, ei, AGG);
    fill_f32<<<cdiv(GG * F, T), T, 0, stream>>>(GSUM, 0.f, GG * F);
    fill_f32<<<cdiv(GG * F, T), T, 0, stream>>>(GVAR, 0.f, GG * F);
    gn_sum<<<cdiv(NN * F, T), T, 0, stream>>>(AGG, batch, GSUM, F);
    gn_center<<<cdiv(NN * F, T), T, 0, stream>>>(AGG, batch, GSUM, CNT, gn2ms, TMP, GVAR, F);
    gn_final<<<cdiv(NN * F, T), T, 0, stream>>>(TMP, batch, GVAR, CNT, gn2w, gn2b, HIO, F);
  }

  // ===== Layer 3: 128 -> 8x8, concat=False (head mean -> F=8) =====
  {
    const int F = 64, C = 8;
    int gb = cdiv((NN / 16) * (F / 16), 8);
    gemm_bias_wmma<128, 64><<<gb, T, 0, stream>>>(HIO, W3l, b3l, XL);
    gemm_bias_wmma<128, 64><<<gb, T, 0, stream>>>(HIO, W3r, b3r, XR);
    fill_u32<<<cdiv(NN * 8, T), T, 0, stream>>>(AMAX, 0u, NN * 8);
    fill_f32<<<cdiv(NN * 8, T), T, 0, stream>>>(ASUM, 0.f, NN * 8);
    attn_logits<64, 8><<<eb, T, 0, stream>>>(XL, XR, att3, ei, ABUF, AMAX);
    attn_exp<<<ehb, T, 0, stream>>>(ABUF, AMAX, ASUM, ei);
    fill_f32<<<cdiv(NN * F, T), T, 0, stream>>>(AGG, 0.f, NN * F);
    attn_aggregate<64, 8><<<eb, T, 0, stream>>>(XL, ABUF, ASUM, ei, AGG);
    head_mean<<<cdiv(NN * 8, T), T, 0, stream>>>(AGG, bias3, H3);
    // GraphNorm on F=8
    fill_f32<<<1, GG * 8, 0, stream>>>(GSUM, 0.f, GG * 8);
    fill_f32<<<1, GG * 8, 0, stream>>>(GVAR, 0.f, GG * 8);
    gn_sum<<<cdiv(NN * 8, T), T, 0, stream>>>(H3, batch, GSUM, 8);
    gn_center<<<cdiv(NN * 8, T), T, 0, stream>>>(H3, batch, GSUM, CNT, gn3ms, TMP, GVAR, 8);
    gn_final<<<cdiv(NN * 8, T), T, 0, stream>>>(TMP, batch, GVAR, CNT, gn3w, gn3b, H3GN, 8);
  }

  // ===== mean pool + linear head =====
  fill_f32<<<1, GG * 8, 0, stream>>>(FEAT, 0.f, GG * 8);
  pool_sum<<<cdiv(NN * 8, T), T, 0, stream>>>(H3GN, batch, FEAT);
  pool_final<<<1, GG, 0, stream>>>(FEAT, CNT, lin_w, lin_b, out);
}